// MNISTMemoryEngine_54752243090068
// MI455X (gfx1250) — compile-verified
//
#include <hip/hip_runtime.h>
#include <math.h>

// ---------------------------------------------------------------------------
// MI455X (gfx1250) implementation of the hierarchical MemoryEngine reference.
// All GEMMs use the fp32 WMMA V_WMMA_F32_16X16X4_F32 (wave32, 16x16 C tile,
// K walked in chunks of 4).  fp32 is chosen deliberately: the 128-step gated
// recurrence is numerically sensitive and the reference is fp32 end-to-end.
// ---------------------------------------------------------------------------

typedef float v2f __attribute__((ext_vector_type(2)));
typedef float v8f __attribute__((ext_vector_type(8)));

#define ZACC {0.f,0.f,0.f,0.f,0.f,0.f,0.f,0.f}

static __device__ __forceinline__ v8f wmma4(float a0, float a1, float b0, float b1, v8f c) {
  v2f a; a[0] = a0; a[1] = a1;
  v2f b; b[0] = b0; b[1] = b1;
  // D = A(16x4) * B(4x16) + C(16x16), fp32
  return __builtin_amdgcn_wmma_f32_16x16x4_f32(false, a, false, b, (short)0, c, false, false);
}

static __device__ __forceinline__ float sigmoidf_(float x) {
  return 1.0f / (1.0f + expf(-x));
}

// Problem constants
#define Bsz 256
#define NPAT 1024
#define LD 128
#define MD 256
#define HD 512
#define NL 8
#define NM 2
#define TLOW 128

// ---------------------------------------------------------------------------
// K0: Wcomb[n] = W_adapt[n] (128x128) @ low_Win[n] (128x128)
// grid = 8 (n), block = 256 (8 waves); each wave: col tile w, loops 8 row tiles
// ---------------------------------------------------------------------------
__global__ void k_wcomb(const float* __restrict__ Wa, const float* __restrict__ Win,
                        float* __restrict__ wcomb) {
  const int n = blockIdx.x, tid = threadIdx.x;
  const int lane = tid & 31, lo = lane & 15, hi = lane >> 4, w = tid >> 5;
  const float* A = Wa + n * 16384;
  const float* B = Win + n * 16384;
  float* C = wcomb + n * 16384;
  const int col = w * 16 + lo;
  for (int rt = 0; rt < 8; ++rt) {
    v8f acc = ZACC;
    for (int kb = 0; kb < 128; kb += 4) {
      const int k0 = kb + hi * 2;
      acc = wmma4(A[(rt * 16 + lo) * 128 + k0], A[(rt * 16 + lo) * 128 + k0 + 1],
                  B[k0 * 128 + col], B[(k0 + 1) * 128 + col], acc);
    }
    for (int r = 0; r < 8; ++r)
      C[(rt * 16 + r + hi * 8) * 128 + col] = acc[r];
  }
}

// ---------------------------------------------------------------------------
// K1 (fused): patchify -> @W_patch(+pad K 49->52) -> +pos -> LN -> @Wcomb[n]
//             -> u_low[n][b][t][d]
// grid = B*NL*8 (b, n, ttile of 16 t-rows), block 256
// ---------------------------------------------------------------------------
__global__ void k_ulow(const float* __restrict__ images, const float* __restrict__ Wp,
                       const float* __restrict__ ln1g, const float* __restrict__ ln1b,
                       const float* __restrict__ pos, const float* __restrict__ wcomb,
                       float* __restrict__ ulow) {
  __shared__ float pt[16][52];   // patch tile (K padded)
  __shared__ float bt[16][129];  // base tile (odd stride -> no LDS bank conflict)
  const int tid = threadIdx.x;
  const int ttile = blockIdx.x & 7;
  const int n = (blockIdx.x >> 3) & 7;
  const int b = blockIdx.x >> 6;

  for (int idx = tid; idx < 16 * 52; idx += 256) {
    const int j = idx / 52, i = idx % 52;
    float v = 0.f;
    if (i < 49) {
      const int t = ttile * 16 + j;
      const int p = n * 128 + t;
      const int gy = p >> 5, gx = p & 31;
      const int py = i / 7, px = i % 7;
      v = images[(b * 224 + gy * 7 + py) * 224 + gx * 7 + px];
    }
    pt[j][i] = v;
  }
  __syncthreads();

  const int lane = tid & 31, lo = lane & 15, hi = lane >> 4, w = tid >> 5;
  const int col = w * 16 + lo;

  // stage 1: patches @ W_patch (K = 49 padded to 52)
  v8f acc = ZACC;
  for (int kb = 0; kb < 52; kb += 4) {
    const int k0 = kb + hi * 2, k1 = k0 + 1;
    const float b0 = (k0 < 49) ? Wp[k0 * 128 + col] : 0.f;
    const float b1 = (k1 < 49) ? Wp[k1 * 128 + col] : 0.f;
    acc = wmma4(pt[lo][k0], pt[lo][k1], b0, b1, acc);
  }
  for (int r = 0; r < 8; ++r) {
    const int rowj = r + hi * 8;
    const int p = n * 128 + ttile * 16 + rowj;
    bt[rowj][col] = acc[r] + pos[p * 128 + col];
  }
  __syncthreads();

  // stage 2: LayerNorm per row (16 rows, one thread each)
  if (tid < 16) {
    float mu = 0.f;
    for (int d = 0; d < 128; ++d) mu += bt[tid][d];
    mu *= (1.f / 128.f);
    float var = 0.f;
    for (int d = 0; d < 128; ++d) { const float x = bt[tid][d] - mu; var += x * x; }
    var *= (1.f / 128.f);
    const float rs = rsqrtf(var + 1e-5f);
    for (int d = 0; d < 128; ++d)
      bt[tid][d] = (bt[tid][d] - mu) * rs * ln1g[d] + ln1b[d];
  }
  __syncthreads();

  // stage 3: base @ Wcomb[n] -> u_low
  const float* Wc = wcomb + n * 16384;
  v8f acc2 = ZACC;
  for (int kb = 0; kb < 128; kb += 4) {
    const int k0 = kb + hi * 2;
    acc2 = wmma4(bt[lo][k0], bt[lo][k0 + 1], Wc[k0 * 128 + col], Wc[(k0 + 1) * 128 + col], acc2);
  }
  for (int r = 0; r < 8; ++r) {
    const int t = ttile * 16 + r + hi * 8;
    ulow[((size_t)(n * Bsz + b) * TLOW + t) * 128 + col] = acc2[r];
  }
}

// ---------------------------------------------------------------------------
// K2: low-level scan. grid = NL*16 (node, 16-batch tile), block 256 (8 waves).
// h ping-synced through LDS; m in registers; Wrec streamed from L2 (64KB/node).
// ---------------------------------------------------------------------------
__global__ void k_lowscan(const float* __restrict__ ulow, const float* __restrict__ Wrec,
                          float* __restrict__ hl, float* __restrict__ ml) {
  __shared__ float hbuf[16][129];
  __shared__ float resp[8][16];
  const int tid = threadIdx.x;
  const int n = blockIdx.x >> 4;
  const int b0 = (blockIdx.x & 15) * 16;
  const int lane = tid & 31, lo = lane & 15, hi = lane >> 4, w = tid >> 5;
  const int col = w * 16 + lo;
  const float* Wr = Wrec + n * 16384;
  const float* ub = ulow + (size_t)(n * Bsz + b0) * TLOW * 128;

  float* hb = &hbuf[0][0];
  for (int idx = tid; idx < 16 * 129; idx += 256) hb[idx] = 0.f;
  float m[8], h[8];
  for (int r = 0; r < 8; ++r) m[r] = 0.f;
  __syncthreads();

  for (int t = 0; t < TLOW; ++t) {
    v8f acc = ZACC;
    for (int kb = 0; kb < 128; kb += 4) {
      const int k0 = kb + hi * 2;
      acc = wmma4(hbuf[lo][k0], hbuf[lo][k0 + 1], Wr[k0 * 128 + col], Wr[(k0 + 1) * 128 + col], acc);
    }
    for (int r = 0; r < 8; ++r) {
      const int row = r + hi * 8;
      const float u = ub[(size_t)row * (TLOW * 128) + t * 128 + col];
      h[r] = tanhf(u + 0.10f * acc[r]);
    }
    // resonance partial: per-row dot(h, m) over this wave's 16 columns
    for (int r = 0; r < 8; ++r) {
      float p = h[r] * m[r];
      for (int off = 1; off < 16; off <<= 1) p += __shfl_xor(p, off, 32);
      if (lo == 0) resp[w][r + hi * 8] = p;
    }
    __syncthreads();
    for (int r = 0; r < 8; ++r) {
      const int row = r + hi * 8;
      float tot = 0.f;
      for (int ww = 0; ww < 8; ++ww) tot += resp[ww][row];
      const float res = sigmoidf_(tot * 0.08838834764831845f - 0.15f);
      m[r] = 0.90f * m[r] + 0.10f * res * h[r];
      hbuf[row][col] = h[r];
    }
    __syncthreads();
  }
  for (int r = 0; r < 8; ++r) {
    const int row = r + hi * 8;
    hl[(n * Bsz + b0 + row) * 128 + col] = h[r];
    ml[(n * Bsz + b0 + row) * 128 + col] = m[r];
  }
}

// ---------------------------------------------------------------------------
// K3: low_msgs[b][n][:] = concat(h_l,m_l)[n,b] (K=256) @ W_l2m[n] (256x256)
// grid = NL*16, block 256; each wave does 2 column tiles.
// ---------------------------------------------------------------------------
__global__ void k_l2m(const float* __restrict__ hl, const float* __restrict__ ml,
                      const float* __restrict__ W, float* __restrict__ msgs) {
  __shared__ float at[16][257];
  const int tid = threadIdx.x;
  const int n = blockIdx.x >> 4;
  const int b0 = (blockIdx.x & 15) * 16;
  for (int idx = tid; idx < 16 * 256; idx += 256) {
    const int j = idx >> 8, k = idx & 255;
    at[j][k] = (k < 128) ? hl[(n * Bsz + b0 + j) * 128 + k]
                         : ml[(n * Bsz + b0 + j) * 128 + (k - 128)];
  }
  __syncthreads();
  const int lane = tid & 31, lo = lane & 15, hi = lane >> 4, w = tid >> 5;
  const float* Wn = W + n * 65536;
  for (int cti = 0; cti < 2; ++cti) {
    const int col = (w + cti * 8) * 16 + lo;
    v8f acc = ZACC;
    for (int kb = 0; kb < 256; kb += 4) {
      const int k0 = kb + hi * 2;
      acc = wmma4(at[lo][k0], at[lo][k0 + 1], Wn[k0 * 256 + col], Wn[(k0 + 1) * 256 + col], acc);
    }
    for (int r = 0; r < 8; ++r)
      msgs[((b0 + r + hi * 8) * NL + n) * 256 + col] = acc[r];
  }
}

// ---------------------------------------------------------------------------
// K4: u_mid[nm][b][g][:] = low_msgs[b][nm*4+g][:] @ mid_Win[nm]
// grid = 16(btile)*2(nm)*4(g) = 128
// ---------------------------------------------------------------------------
__global__ void k_midu(const float* __restrict__ msgs, const float* __restrict__ Win,
                       float* __restrict__ umid) {
  __shared__ float at[16][257];
  const int tid = threadIdx.x;
  const int g = blockIdx.x & 3;
  const int nm = (blockIdx.x >> 2) & 1;
  const int b0 = (blockIdx.x >> 3) * 16;
  for (int idx = tid; idx < 16 * 256; idx += 256) {
    const int j = idx >> 8, k = idx & 255;
    at[j][k] = msgs[((b0 + j) * NL + nm * 4 + g) * 256 + k];
  }
  __syncthreads();
  const int lane = tid & 31, lo = lane & 15, hi = lane >> 4, w = tid >> 5;
  const float* Wn = Win + nm * 65536;
  for (int cti = 0; cti < 2; ++cti) {
    const int col = (w + cti * 8) * 16 + lo;
    v8f acc = ZACC;
    for (int kb = 0; kb < 256; kb += 4) {
      const int k0 = kb + hi * 2;
      acc = wmma4(at[lo][k0], at[lo][k0 + 1], Wn[k0 * 256 + col], Wn[(k0 + 1) * 256 + col], acc);
    }
    for (int r = 0; r < 8; ++r)
      umid[((nm * Bsz + b0 + r + hi * 8) * 4 + 0) * 256 + g * 256 + col] = acc[r];
  }
}

// ---------------------------------------------------------------------------
// K5: mid-level scan (T=4, D=256). grid = NM*16, block 256.
// Each wave owns 2 column tiles.
// ---------------------------------------------------------------------------
__global__ void k_midscan(const float* __restrict__ umid, const float* __restrict__ Wrec,
                          float* __restrict__ hm, float* __restrict__ mm) {
  __shared__ float hbuf[16][257];
  __shared__ float resp[8][16];
  const int tid = threadIdx.x;
  const int nm = blockIdx.x >> 4;
  const int b0 = (blockIdx.x & 15) * 16;
  const int lane = tid & 31, lo = lane & 15, hi = lane >> 4, w = tid >> 5;
  const float* Wr = Wrec + nm * 65536;

  float* hb = &hbuf[0][0];
  for (int idx = tid; idx < 16 * 257; idx += 256) hb[idx] = 0.f;
  float m[2][8], h[2][8];
  for (int c = 0; c < 2; ++c)
    for (int r = 0; r < 8; ++r) m[c][r] = 0.f;
  __syncthreads();

  for (int t = 0; t < 4; ++t) {
    for (int cti = 0; cti < 2; ++cti) {
      const int col = (w + cti * 8) * 16 + lo;
      v8f acc = ZACC;
      for (int kb = 0; kb < 256; kb += 4) {
        const int k0 = kb + hi * 2;
        acc = wmma4(hbuf[lo][k0], hbuf[lo][k0 + 1], Wr[k0 * 256 + col], Wr[(k0 + 1) * 256 + col], acc);
      }
      for (int r = 0; r < 8; ++r) {
        const int row = r + hi * 8;
        const float u = umid[((nm * Bsz + b0 + row) * 4 + t) * 256 + col];
        h[cti][r] = tanhf(u + 0.12f * acc[r]);
      }
    }
    for (int r = 0; r < 8; ++r) {
      float p = h[0][r] * m[0][r] + h[1][r] * m[1][r];
      for (int off = 1; off < 16; off <<= 1) p += __shfl_xor(p, off, 32);
      if (lo == 0) resp[w][r + hi * 8] = p;
    }
    __syncthreads();
    for (int r = 0; r < 8; ++r) {
      const int row = r + hi * 8;
      float tot = 0.f;
      for (int ww = 0; ww < 8; ++ww) tot += resp[ww][row];
      const float res = sigmoidf_(tot * 0.0625f - 0.18f);
      for (int cti = 0; cti < 2; ++cti) {
        m[cti][r] = 0.92f * m[cti][r] + 0.08f * res * h[cti][r];
        hbuf[row][(w + cti * 8) * 16 + lo] = h[cti][r];
      }
    }
    __syncthreads();
  }
  for (int cti = 0; cti < 2; ++cti) {
    const int col = (w + cti * 8) * 16 + lo;
    for (int r = 0; r < 8; ++r) {
      const int row = r + hi * 8;
      hm[(nm * Bsz + b0 + row) * 256 + col] = h[cti][r];
      mm[(nm * Bsz + b0 + row) * 256 + col] = m[cti][r];
    }
  }
}

// ---------------------------------------------------------------------------
// K6: high_in[b][nm][:] = concat(h_m,m_m)[nm,b] (K=512) @ W_m2h[nm] (512x512)
// grid = NM*16; each wave 4 column tiles.
// ---------------------------------------------------------------------------
__global__ void k_m2h(const float* __restrict__ hm, const float* __restrict__ mm,
                      const float* __restrict__ W, float* __restrict__ hin) {
  __shared__ float at[16][513];
  const int tid = threadIdx.x;
  const int nm = blockIdx.x >> 4;
  const int b0 = (blockIdx.x & 15) * 16;
  for (int idx = tid; idx < 16 * 512; idx += 256) {
    const int j = idx >> 9, k = idx & 511;
    at[j][k] = (k < 256) ? hm[(nm * Bsz + b0 + j) * 256 + k]
                         : mm[(nm * Bsz + b0 + j) * 256 + (k - 256)];
  }
  __syncthreads();
  const int lane = tid & 31, lo = lane & 15, hi = lane >> 4, w = tid >> 5;
  const float* Wn = W + nm * 262144;
  for (int cti = 0; cti < 4; ++cti) {
    const int col = (w + cti * 8) * 16 + lo;
    v8f acc = ZACC;
    for (int kb = 0; kb < 512; kb += 4) {
      const int k0 = kb + hi * 2;
      acc = wmma4(at[lo][k0], at[lo][k0 + 1], Wn[k0 * 512 + col], Wn[(k0 + 1) * 512 + col], acc);
    }
    for (int r = 0; r < 8; ++r)
      hin[((b0 + r + hi * 8) * NM + nm) * 512 + col] = acc[r];
  }
}

// ---------------------------------------------------------------------------
// K7: u_high[t][b][:] = high_in[b][t][:] @ high_Win (512x512). grid = 2*16.
// ---------------------------------------------------------------------------
__global__ void k_highu(const float* __restrict__ hin, const float* __restrict__ Win,
                        float* __restrict__ uhigh) {
  __shared__ float at[16][513];
  const int tid = threadIdx.x;
  const int t = blockIdx.x >> 4;
  const int b0 = (blockIdx.x & 15) * 16;
  for (int idx = tid; idx < 16 * 512; idx += 256) {
    const int j = idx >> 9, k = idx & 511;
    at[j][k] = hin[((b0 + j) * NM + t) * 512 + k];
  }
  __syncthreads();
  const int lane = tid & 31, lo = lane & 15, hi = lane >> 4, w = tid >> 5;
  for (int cti = 0; cti < 4; ++cti) {
    const int col = (w + cti * 8) * 16 + lo;
    v8f acc = ZACC;
    for (int kb = 0; kb < 512; kb += 4) {
      const int k0 = kb + hi * 2;
      acc = wmma4(at[lo][k0], at[lo][k0 + 1], Win[k0 * 512 + col], Win[(k0 + 1) * 512 + col], acc);
    }
    for (int r = 0; r < 8; ++r)
      uhigh[(size_t)(t * Bsz + b0 + r + hi * 8) * 512 + col] = acc[r];
  }
}

// ---------------------------------------------------------------------------
// K8: high node (T=2, delay=2 => recurrence term exactly 0), LN(1024), head.
// grid = B, block 256 (each thread handles 2 of the 512 dims).
// ---------------------------------------------------------------------------
static __device__ float block_reduce(float v, float* red, int tid) {
  __syncthreads();
  red[tid] = v;
  __syncthreads();
  for (int s = 128; s > 0; s >>= 1) {
    if (tid < s) red[tid] += red[tid + s];
    __syncthreads();
  }
  return red[0];
}

__global__ void k_final(const float* __restrict__ uhigh, const float* __restrict__ ln2g,
                        const float* __restrict__ ln2b, const float* __restrict__ Wcls,
                        const float* __restrict__ bcls, float* __restrict__ out) {
  __shared__ float tape[1024];
  __shared__ float red[256];
  const int b = blockIdx.x, tid = threadIdx.x;
  const float res1 = 1.f / (1.f + expf(0.20f));  // sigmoid(0 - theta)

  float h2v[2], m1v[2];
  float pdot = 0.f;
  for (int i = 0; i < 2; ++i) {
    const int d = tid + i * 256;
    const float h1 = tanhf(uhigh[(size_t)b * 512 + d]);
    const float m1 = 0.06f * res1 * h1;
    const float h2 = tanhf(uhigh[(size_t)(Bsz + b) * 512 + d]);
    pdot += h2 * m1;
    h2v[i] = h2;
    m1v[i] = m1;
  }
  const float dot = block_reduce(pdot, red, tid);
  const float res2 = sigmoidf_(dot * 0.04419417382415922f - 0.20f);
  for (int i = 0; i < 2; ++i) {
    const int d = tid + i * 256;
    const float m2 = 0.94f * m1v[i] + 0.06f * res2 * h2v[i];
    tape[d] = h2v[i];
    tape[512 + d] = m2;
  }
  __syncthreads();

  float s1 = 0.f, s2 = 0.f;
  for (int i = 0; i < 4; ++i) {
    const float x = tape[tid + i * 256];
    s1 += x;
    s2 += x * x;
  }
  const float mu = block_reduce(s1, red, tid) * (1.f / 1024.f);
  const float ex2 = block_reduce(s2, red, tid) * (1.f / 1024.f);
  const float rs = rsqrtf(ex2 - mu * mu + 1e-5f);

  float pc[10];
  for (int c = 0; c < 10; ++c) pc[c] = 0.f;
  for (int i = 0; i < 4; ++i) {
    const int d = tid + i * 256;
    const float xn = (tape[d] - mu) * rs * ln2g[d] + ln2b[d];
    for (int c = 0; c < 10; ++c) pc[c] += xn * Wcls[d * 10 + c];
  }
  for (int c = 0; c < 10; ++c) {
    const float s = block_reduce(pc[c], red, tid);
    if (tid == 0) out[b * 10 + c] = s + bcls[c];
  }
}

// ---------------------------------------------------------------------------
// Launch
// ---------------------------------------------------------------------------
extern "C" void kernel_launch(void* const* d_in, const int* in_sizes, int n_in,
                              void* d_out, int out_size, void* d_ws, size_t ws_size,
                              hipStream_t stream) {
  (void)in_sizes; (void)n_in; (void)out_size; (void)ws_size;
  const float* images   = (const float*)d_in[0];
  const float* W_patch  = (const float*)d_in[1];
  const float* ln1_g    = (const float*)d_in[2];
  const float* ln1_b    = (const float*)d_in[3];
  const float* pos_emb  = (const float*)d_in[4];
  const float* W_adapt  = (const float*)d_in[5];
  const float* low_Win  = (const float*)d_in[6];
  const float* low_Wrec = (const float*)d_in[7];
  const float* W_l2m    = (const float*)d_in[8];
  const float* mid_Win  = (const float*)d_in[9];
  const float* mid_Wrec = (const float*)d_in[10];
  const float* W_m2h    = (const float*)d_in[11];
  const float* high_Win = (const float*)d_in[12];
  // d_in[13] = high_Wrec: with T=2, delay=2 the recurrence term is exactly 0.
  const float* ln2_g    = (const float*)d_in[14];
  const float* ln2_b    = (const float*)d_in[15];
  const float* W_cls    = (const float*)d_in[16];
  const float* b_cls    = (const float*)d_in[17];
  float* out = (float*)d_out;
  float* ws = (float*)d_ws;

  // workspace layout (floats)
  float* wcomb = ws;                       // 8*128*128      = 131072
  float* ulow  = wcomb + 131072;           // 8*256*128*128  = 33554432
  float* hl    = ulow  + 33554432;         // 8*256*128      = 262144
  float* ml    = hl    + 262144;           // 262144
  float* msgs  = ml    + 262144;           // 256*8*256      = 524288
  float* umid  = msgs  + 524288;           // 2*256*4*256    = 524288
  float* hm    = umid  + 524288;           // 2*256*256      = 131072
  float* mm    = hm    + 131072;           // 131072
  float* hin   = mm    + 131072;           // 256*2*512      = 262144
  float* uhigh = hin   + 262144;           // 2*256*512      = 262144

  dim3 blk(256);
  k_wcomb  <<<dim3(8),            blk, 0, stream>>>(W_adapt, low_Win, wcomb);
  k_ulow   <<<dim3(Bsz * NL * 8), blk, 0, stream>>>(images, W_patch, ln1_g, ln1_b,
                                                    pos_emb, wcomb, ulow);
  k_lowscan<<<dim3(NL * 16),      blk, 0, stream>>>(ulow, low_Wrec, hl, ml);
  k_l2m    <<<dim3(NL * 16),      blk, 0, stream>>>(hl, ml, W_l2m, msgs);
  k_midu   <<<dim3(128),          blk, 0, stream>>>(msgs, mid_Win, umid);
  k_midscan<<<dim3(NM * 16),      blk, 0, stream>>>(umid, mid_Wrec, hm, mm);
  k_m2h    <<<dim3(NM * 16),      blk, 0, stream>>>(hm, mm, W_m2h, hin);
  k_highu  <<<dim3(NM * 16),      blk, 0, stream>>>(hin, high_Win, uhigh);
  k_final  <<<dim3(Bsz),          blk, 0, stream>>>(uhigh, ln2_g, ln2_b, W_cls, b_cls, out);
}